// SigmoidClipLoss_54082228191242
// MI455X (gfx1250) — compile-verified
//
#include <hip/hip_runtime.h>
#include <math.h>

#define BSZ 8192
#define DIM 256
#define TILE 128
#define KCHUNK 16
#define LSTRIDE (KCHUNK + 4)       // 20 floats: row bank-shift of 4 -> conflict-free b64 frag loads
#define NCHUNK (DIM / KCHUNK)      // 16

typedef float v2f __attribute__((ext_vector_type(2)));
typedef float v8f __attribute__((ext_vector_type(8)));
typedef int   v4i_t __attribute__((vector_size(16)));

#define AS1 __attribute__((address_space(1)))
#define AS3 __attribute__((address_space(3)))

#if defined(__HIP_DEVICE_COMPILE__) && __has_builtin(__builtin_amdgcn_global_load_async_to_lds_b128)
#define HAVE_ASYNC 1
#else
#define HAVE_ASYNC 0
#endif

// ---------------------------------------------------------------------------
// Kernel 1: L2-normalize rows of both embedding matrices into workspace.
// ---------------------------------------------------------------------------
__global__ __launch_bounds__(128) void siglip_normalize_kernel(
    const float* __restrict__ loc, const float* __restrict__ che,
    float* __restrict__ locN, float* __restrict__ cheN)
{
    const int wave = threadIdx.x >> 5;
    const int lane = threadIdx.x & 31;
    const int row  = blockIdx.x * 4 + wave;   // 0..16383

    const float* src;
    float*       dst;
    if (row < BSZ) {
        src = che + (size_t)row * DIM;
        dst = cheN + (size_t)row * DIM;
    } else {
        src = loc + (size_t)(row - BSZ) * DIM;
        dst = locN + (size_t)(row - BSZ) * DIM;
    }

    float4 a = ((const float4*)src)[lane];
    float4 b = ((const float4*)src)[lane + 32];

    float ss = a.x * a.x + a.y * a.y + a.z * a.z + a.w * a.w
             + b.x * b.x + b.y * b.y + b.z * b.z + b.w * b.w;
    #pragma unroll
    for (int off = 16; off > 0; off >>= 1)
        ss += __shfl_xor(ss, off, 32);

    const float inv = 1.0f / sqrtf(ss);
    a.x *= inv; a.y *= inv; a.z *= inv; a.w *= inv;
    b.x *= inv; b.y *= inv; b.z *= inv; b.w *= inv;
    ((float4*)dst)[lane]      = a;
    ((float4*)dst)[lane + 32] = b;
}

// ---------------------------------------------------------------------------
// Async global -> LDS 16-byte copy (ASYNCcnt-tracked, bypasses VGPRs).
// Builtin signature on this toolchain: (v4i as1*, v4i as3*, imm offset, imm cpol).
// Generic LDS pointer's low 32 bits are the LDS byte offset (ISA aperture rule).
// ---------------------------------------------------------------------------
__device__ __forceinline__ void async_copy16(const float* g, float* l) {
#if HAVE_ASYNC
    __builtin_amdgcn_global_load_async_to_lds_b128(
        (AS1 v4i_t*)(uintptr_t)g,
        (AS3 v4i_t*)(unsigned)(uintptr_t)l,
        0, 0);
#else
    (void)g; (void)l;
#endif
}

__device__ __forceinline__ void async_wait0() {
#if HAVE_ASYNC
#if __has_builtin(__builtin_amdgcn_s_wait_asynccnt)
    __builtin_amdgcn_s_wait_asynccnt(0);
#else
    asm volatile("s_wait_asynccnt 0x0" ::: "memory");
#endif
#endif
}

// ---------------------------------------------------------------------------
// Kernel 2: fused GEMM (f32 WMMA 16x16x4) + scale/shift + softplus + reduce.
// Block = 256 threads = 8 waves -> one 128x128 logits tile.
// Wave (wm,wn) = (wave>>1, wave&1) owns a 32x64 subtile: acc[2][4] of v8f.
// K double-buffered in LDS with async global->LDS copies.
// ---------------------------------------------------------------------------
__global__ __launch_bounds__(256) void siglip_loss_kernel(
    const float* __restrict__ cheN, const float* __restrict__ locN,
    const float* __restrict__ t_prime, const float* __restrict__ bptr,
    float* __restrict__ out)
{
    __shared__ float lA[2][TILE * LSTRIDE];   // che tile rows (M / logits i)
    __shared__ float lB[2][TILE * LSTRIDE];   // loc tile rows (N / logits j)
    __shared__ float red[8];

    const int tid  = threadIdx.x;
    const int lane = tid & 31;
    const int wave = tid >> 5;
    const int bm   = blockIdx.x;           // che tile
    const int bn   = blockIdx.y;           // loc tile
    const int wm   = wave >> 1;            // 0..3 : 32-row strip
    const int wn   = wave & 1;             // 0..1 : 64-col strip
    const int half = lane >> 4;            // 0/1  : K-half of fragment
    const int l16  = lane & 15;            // row-within-16 / N-within-16

    v8f acc[2][4];
    #pragma unroll
    for (int im = 0; im < 2; ++im)
        #pragma unroll
        for (int in = 0; in < 4; ++in)
            acc[im][in] = (v8f){0.f, 0.f, 0.f, 0.f, 0.f, 0.f, 0.f, 0.f};

    const size_t abase = (size_t)(bm * TILE) * DIM;
    const size_t bbase = (size_t)(bn * TILE) * DIM;

    // Per-chunk cooperative copy: 128 rows x 16 cols = 512 float4 / 256 threads.
    const int r0 = tid >> 2;               // thread's first row   (0..63)
    const int c0 = (tid & 3) << 2;         // thread's float4 col  (0,4,8,12)

#if HAVE_ASYNC
    // ---- software-pipelined, double-buffered, async global->LDS ----
    {
        #pragma unroll
        for (int i = 0; i < 2; ++i) {
            const int r = r0 + i * 64;
            async_copy16(cheN + abase + (size_t)r * DIM + c0, &lA[0][r * LSTRIDE + c0]);
            async_copy16(locN + bbase + (size_t)r * DIM + c0, &lB[0][r * LSTRIDE + c0]);
        }
    }
    for (int c = 0; c < NCHUNK; ++c) {
        const int cur = c & 1;
        async_wait0();                     // this wave's async copies landed
        __syncthreads();                   // everyone's copies landed; prev reads done

        if (c + 1 < NCHUNK) {              // prefetch next chunk into alternate buffer
            const int k1 = (c + 1) * KCHUNK;
            #pragma unroll
            for (int i = 0; i < 2; ++i) {
                const int r = r0 + i * 64;
                async_copy16(cheN + abase + (size_t)r * DIM + k1 + c0, &lA[cur ^ 1][r * LSTRIDE + c0]);
                async_copy16(locN + bbase + (size_t)r * DIM + k1 + c0, &lB[cur ^ 1][r * LSTRIDE + c0]);
            }
        }

        #pragma unroll
        for (int kk = 0; kk < KCHUNK; kk += 4) {
            // ISA 16x4 f32 A layout: lane%16 = row, K = vgpr + 2*(lane>=16).
            v2f afrag[2], bfrag[4];
            #pragma unroll
            for (int im = 0; im < 2; ++im)
                afrag[im] = *(const v2f*)&lA[cur][(wm * 32 + im * 16 + l16) * LSTRIDE + kk + 2 * half];
            #pragma unroll
            for (int in = 0; in < 4; ++in)
                bfrag[in] = *(const v2f*)&lB[cur][(wn * 64 + in * 16 + l16) * LSTRIDE + kk + 2 * half];

            #pragma unroll
            for (int im = 0; im < 2; ++im)
                #pragma unroll
                for (int in = 0; in < 4; ++in)
                    acc[im][in] = __builtin_amdgcn_wmma_f32_16x16x4_f32(
                        false, afrag[im], false, bfrag[in],
                        (short)0, acc[im][in], false, false);
        }
    }
#else
    // ---- fallback: synchronous staging ----
    for (int c = 0; c < NCHUNK; ++c) {
        const int k0 = c * KCHUNK;
        __syncthreads();
        #pragma unroll
        for (int i = 0; i < 2; ++i) {
            const int r = r0 + i * 64;
            float4 va = *(const float4*)(cheN + abase + (size_t)r * DIM + k0 + c0);
            float4 vb = *(const float4*)(locN + bbase + (size_t)r * DIM + k0 + c0);
            *(float4*)&lA[0][r * LSTRIDE + c0] = va;
            *(float4*)&lB[0][r * LSTRIDE + c0] = vb;
        }
        __syncthreads();
        #pragma unroll
        for (int kk = 0; kk < KCHUNK; kk += 4) {
            v2f afrag[2], bfrag[4];
            #pragma unroll
            for (int im = 0; im < 2; ++im)
                afrag[im] = *(const v2f*)&lA[0][(wm * 32 + im * 16 + l16) * LSTRIDE + kk + 2 * half];
            #pragma unroll
            for (int in = 0; in < 4; ++in)
                bfrag[in] = *(const v2f*)&lB[0][(wn * 64 + in * 16 + l16) * LSTRIDE + kk + 2 * half];
            #pragma unroll
            for (int im = 0; im < 2; ++im)
                #pragma unroll
                for (int in = 0; in < 4; ++in)
                    acc[im][in] = __builtin_amdgcn_wmma_f32_16x16x4_f32(
                        false, afrag[im], false, bfrag[in],
                        (short)0, acc[im][in], false, false);
        }
    }
#endif

    // Epilogue: z = acc*exp(t') + b ; loss += softplus(-y*z), y = +1 on diagonal.
    const float scale = __expf(t_prime[0]);
    const float bias  = bptr[0];
    float lsum = 0.f;

    #pragma unroll
    for (int im = 0; im < 2; ++im) {
        #pragma unroll
        for (int in = 0; in < 4; ++in) {
            #pragma unroll
            for (int j = 0; j < 8; ++j) {
                const float z  = acc[im][in][j] * scale + bias;
                const int   gm = bm * TILE + wm * 32 + im * 16 + j + 8 * half;
                const int   gn = bn * TILE + wn * 64 + in * 16 + l16;
                const float y  = (gm == gn) ? 1.f : -1.f;
                const float x  = -y * z;                  // softplus(x) = -log_sigmoid(y*z)
                const float sp = fmaxf(x, 0.f) + log1pf(__expf(-fabsf(x)));
                lsum += sp;
            }
        }
    }

    #pragma unroll
    for (int off = 16; off > 0; off >>= 1)
        lsum += __shfl_xor(lsum, off, 32);
    if (lane == 0) red[wave] = lsum;
    __syncthreads();
    if (tid == 0) {
        float s = 0.f;
        #pragma unroll
        for (int w = 0; w < 8; ++w) s += red[w];
        atomicAdd(out, s * (1.0f / (float)BSZ));
    }
}

// ---------------------------------------------------------------------------
extern "C" void kernel_launch(void* const* d_in, const int* in_sizes, int n_in,
                              void* d_out, int out_size, void* d_ws, size_t ws_size,
                              hipStream_t stream) {
    const float* loc = (const float*)d_in[0];   // loc_month_emb [B,D]
    const float* che = (const float*)d_in[1];   // chelsa_emb    [B,D]
    const float* tp  = (const float*)d_in[2];   // t_prime       [1]
    const float* bb  = (const float*)d_in[3];   // b             [1]
    float* out = (float*)d_out;

    float* cheN = (float*)d_ws;                      // 8 MB
    float* locN = cheN + (size_t)BSZ * DIM;          // 8 MB

    (void)hipMemsetAsync(out, 0, sizeof(float) * (size_t)out_size, stream);

    siglip_normalize_kernel<<<(2 * BSZ) / 4, 128, 0, stream>>>(loc, che, locN, cheN);

    dim3 grid(BSZ / TILE, BSZ / TILE);   // 64 x 64 tiles
    siglip_loss_kernel<<<grid, 256, 0, stream>>>(cheN, locN, tp, bb, out);
}